// COMPoolNet_global_89060441850433
// MI455X (gfx1250) — compile-verified
//
#include <hip/hip_runtime.h>
#include <hip/hip_bf16.h>
#include <math.h>

typedef __attribute__((ext_vector_type(2))) float v2f;
typedef __attribute__((ext_vector_type(8))) float v8f;

// ---------------------------------------------------------------------------
// 16x16 fp32 WMMA tile:  C += A(16xK) * B(Kx16), K multiple of 4.
// A: row-major, lane l = row M=l; v0/v1 hold K = k+2*half, k+2*half+1.
// B: row-major (pre-offset to column block), rows striped across lanes.
// ---------------------------------------------------------------------------
__device__ __forceinline__ v8f mm16(const float* A, int lda, float rs,
                                    const float* Bm, int ldb, int K,
                                    int half, int l, v8f c) {
    for (int k = 0; k < K; k += 4) {
        v2f a, b;
        size_t ra = (size_t)l * lda + (k + 2 * half);
        a.x = A[ra] * rs;
        a.y = A[ra + 1] * rs;
        size_t rb = (size_t)(k + 2 * half) * ldb + l;
        b.x = Bm[rb];
        b.y = Bm[rb + (size_t)ldb];
        c = __builtin_amdgcn_wmma_f32_16x16x4_f32(false, a, false, b,
                                                  (short)0, c, false, false);
    }
    return c;
}

// Generic GEMM: C[M x ccols] = act(rowscale * A[M x K] @ B[K x N16] + bias)
// One wave per 16x16 tile. grid.x covers M tiles, grid.y covers N tiles.
__global__ void gemm_wmma(const float* __restrict__ A, int lda,
                          const float* __restrict__ Bw, int ldb,
                          const float* __restrict__ bias,
                          const float* __restrict__ rowscale,
                          float* __restrict__ C, int ldc, int ccols,
                          int K, int relu) {
    const int wave = threadIdx.x >> 5;
    const int lane = threadIdx.x & 31;
    const int half = lane >> 4;
    const int l    = lane & 15;
    const int m0 = (blockIdx.x * (blockDim.x >> 5) + wave) * 16;
    const int n0 = blockIdx.y * 16;
    float rs = rowscale ? rowscale[m0 + l] : 1.0f;
    v8f c = {};
    c = mm16(A + (size_t)m0 * lda, lda, rs, Bw + n0, ldb, K, half, l, c);
    const int col = n0 + l;
    float bv = bias[col];
    if (col < ccols) {
        #pragma unroll
        for (int r = 0; r < 8; ++r) {
            float v = c[r] + bv;
            if (relu) v = fmaxf(v, 0.0f);
            C[(size_t)(m0 + r + 8 * half) * ldc + col] = v;
        }
    }
}

// Fused node MLP: out3(N x 384) -> relu@384x192 -> relu@192x96 -> @96x10.
// One block (8 waves) per 16-row tile; intermediates kept in LDS.
__global__ void node_mlp_fused(const float* __restrict__ out3,
                               const float* __restrict__ w0, const float* __restrict__ b0,
                               const float* __restrict__ w1, const float* __restrict__ b1,
                               const float* __restrict__ w2p, const float* __restrict__ b2p,
                               float* __restrict__ pred) {
    __shared__ float s1[16 * 192];
    __shared__ float s2[16 * 96];
    const int wave = threadIdx.x >> 5;
    const int lane = threadIdx.x & 31;
    const int half = lane >> 4;
    const int l    = lane & 15;
    const int m0 = blockIdx.x * 16;

    for (int ct = wave; ct < 12; ct += 8) {           // 192/16 tiles
        v8f c = {};
        c = mm16(out3 + (size_t)m0 * 384, 384, 1.0f, w0 + ct * 16, 192, 384, half, l, c);
        float bv = b0[ct * 16 + l];
        #pragma unroll
        for (int r = 0; r < 8; ++r)
            s1[(r + 8 * half) * 192 + ct * 16 + l] = fmaxf(c[r] + bv, 0.0f);
    }
    __syncthreads();
    for (int ct = wave; ct < 6; ct += 8) {            // 96/16 tiles
        v8f c = {};
        c = mm16(s1, 192, 1.0f, w1 + ct * 16, 96, 192, half, l, c);
        float bv = b1[ct * 16 + l];
        #pragma unroll
        for (int r = 0; r < 8; ++r)
            s2[(r + 8 * half) * 96 + ct * 16 + l] = fmaxf(c[r] + bv, 0.0f);
    }
    __syncthreads();
    if (wave == 0) {                                  // padded 96x16 (10 valid)
        v8f c = {};
        c = mm16(s2, 96, 1.0f, w2p, 16, 96, half, l, c);
        float bv = b2p[l];
        if (l < 10) {
            #pragma unroll
            for (int r = 0; r < 8; ++r)
                pred[(size_t)(m0 + r + 8 * half) * 10 + l] = c[r] + bv;
        }
    }
}

// ------------------------- graph / scatter kernels -------------------------
__global__ void zero_kernel(float* __restrict__ p, size_t n) {
    size_t i = (size_t)blockIdx.x * blockDim.x + threadIdx.x;
    size_t stride = (size_t)gridDim.x * blockDim.x;
    for (; i < n; i += stride) p[i] = 0.0f;
}

__global__ void deg_kernel(const int* __restrict__ src, const int* __restrict__ dst,
                           float* __restrict__ dout, float* __restrict__ din, int E) {
    int e = blockIdx.x * blockDim.x + threadIdx.x;
    if (e < E) {
        atomicAdd(&dout[src[e]], 1.0f);
        atomicAdd(&din[dst[e]], 1.0f);
    }
}

__global__ void norm_kernel(float* __restrict__ d, size_t n) {
    size_t i = (size_t)blockIdx.x * blockDim.x + threadIdx.x;
    if (i < n) d[i] = rsqrtf(fmaxf(d[i], 1.0f));
}

// One wave per edge: agg[dst] += x[src] * norm_out[src]  (128 features)
__global__ void gather_scatter(const float* __restrict__ X, int lda,
                               const int* __restrict__ src, const int* __restrict__ dst,
                               const float* __restrict__ nout,
                               float* __restrict__ agg, int E) {
    int e = blockIdx.x * (blockDim.x >> 5) + (threadIdx.x >> 5);
    if (e >= E) return;
    int lane = threadIdx.x & 31;
    int s = src[e], d = dst[e];
    float ns = nout[s];
    const float* xp = X + (size_t)s * lda;
    float* ap = agg + (size_t)d * 128;
    #pragma unroll
    for (int j = 0; j < 4; ++j)
        atomicAdd(&ap[lane + 32 * j], xp[lane + 32 * j] * ns);
}

// s0 = out3 @ Ws  (384 -> 1), one wave per node.
__global__ void s0_kernel(const float* __restrict__ out3, const float* __restrict__ Ws,
                          float* __restrict__ s0, int N) {
    int node = blockIdx.x * (blockDim.x >> 5) + (threadIdx.x >> 5);
    if (node >= N) return;
    int lane = threadIdx.x & 31;
    const float* p = out3 + (size_t)node * 384;
    float v = 0.0f;
    #pragma unroll
    for (int j = 0; j < 12; ++j) v += p[lane + 32 * j] * Ws[lane + 32 * j];
    for (int off = 16; off > 0; off >>= 1) v += __shfl_xor(v, off, 32);
    if (lane == 0) s0[node] = v;
}

__global__ void score_edge(const int* __restrict__ src, const int* __restrict__ dst,
                           const float* __restrict__ s0, const float* __restrict__ nout,
                           float* __restrict__ sagg, int E) {
    int e = blockIdx.x * blockDim.x + threadIdx.x;
    if (e < E) atomicAdd(&sagg[dst[e]], s0[src[e]] * nout[src[e]]);
}

__global__ void score_fin(const float* __restrict__ sagg, const float* __restrict__ nin,
                          const float* __restrict__ bs, float* __restrict__ score, int N) {
    int i = blockIdx.x * blockDim.x + threadIdx.x;
    if (i < N) score[i] = sagg[i] * nin[i] + bs[0];
}

// Bitonic sort (descending) of 1024 scores per graph; writes local node indices.
__global__ void sort_kernel(const float* __restrict__ score, int* __restrict__ perm) {
    __shared__ float key[1024];
    __shared__ int idx[1024];
    int g = blockIdx.x;
    int t = threadIdx.x;                  // 512 threads
    key[t] = score[g * 1024 + t];           idx[t] = t;
    key[t + 512] = score[g * 1024 + t + 512]; idx[t + 512] = t + 512;
    __syncthreads();
    for (int k = 2; k <= 1024; k <<= 1) {
        for (int j = k >> 1; j > 0; j >>= 1) {
            int i = ((t & ~(j - 1)) << 1) | (t & (j - 1));
            int p = i | j;
            bool desc = ((i & k) == 0);
            float ki = key[i], kp = key[p];
            bool sw = desc ? (ki < kp) : (ki > kp);
            if (sw) {
                key[i] = kp; key[p] = ki;
                int tmp = idx[i]; idx[i] = idx[p]; idx[p] = tmp;
            }
            __syncthreads();
        }
    }
    perm[g * 1024 + t] = idx[t];
    perm[g * 1024 + t + 512] = idx[t + 512];
}

// Pool mean/max over 512 selected nodes, gated by tanh(score).
__global__ void pool_kernel(const float* __restrict__ out3, const float* __restrict__ score,
                            const int* __restrict__ perm,
                            float* __restrict__ hg3, float* __restrict__ hg3_com) {
    __shared__ float tl[512];
    __shared__ int nl[512];
    int g = blockIdx.x;
    int hs = blockIdx.y;                  // 0 = top-k (dis), 1 = bottom (com)
    int tid = threadIdx.x;                // 384 threads
    for (int p = tid; p < 512; p += 384) {
        int node = perm[g * 1024 + hs * 512 + p];
        nl[p] = node;
        tl[p] = tanhf(score[g * 1024 + node]);
    }
    __syncthreads();
    if (tid < 384) {
        float sum = 0.0f, mx = -3.402823466e+38f;
        for (int p = 0; p < 512; ++p) {
            float v = out3[((size_t)g * 1024 + nl[p]) * 384 + tid] * tl[p];
            sum += v;
            mx = fmaxf(mx, v);
        }
        float* o = hs ? hg3_com : hg3;
        o[(size_t)g * 768 + tid] = sum * (1.0f / 512.0f);
        o[(size_t)g * 768 + 384 + tid] = mx;
    }
}

// Global softmax over all N scores (single block).
__global__ void softmax_kernel(const float* __restrict__ score, float* __restrict__ out, int N) {
    __shared__ float red[1024];
    __shared__ float s_max, s_sum;
    int t = threadIdx.x;
    float m = -3.402823466e+38f;
    for (int i = t; i < N; i += 1024) m = fmaxf(m, score[i]);
    red[t] = m; __syncthreads();
    for (int s = 512; s > 0; s >>= 1) { if (t < s) red[t] = fmaxf(red[t], red[t + s]); __syncthreads(); }
    if (t == 0) s_max = red[0];
    __syncthreads();
    float sum = 0.0f;
    for (int i = t; i < N; i += 1024) sum += expf(score[i] - s_max);
    red[t] = sum; __syncthreads();
    for (int s = 512; s > 0; s >>= 1) { if (t < s) red[t] += red[t + s]; __syncthreads(); }
    if (t == 0) s_sum = red[0];
    __syncthreads();
    for (int i = t; i < N; i += 1024) out[i] = expf(score[i] - s_max) / s_sum;
}

// Pad the 10-column classifier weights to 16 columns (zeros).
__global__ void pad_kernel(const float* __restrict__ mn_w2, const float* __restrict__ mn_b2,
                           const float* __restrict__ mg_w2, const float* __restrict__ mg_b2,
                           float* __restrict__ w2p_n, float* __restrict__ b2p_n,
                           float* __restrict__ w2p_g, float* __restrict__ b2p_g) {
    int i = blockIdx.x * blockDim.x + threadIdx.x;
    if (i < 96 * 16)  { int r = i >> 4, c = i & 15; w2p_n[i] = (c < 10) ? mn_w2[r * 10 + c] : 0.0f; }
    if (i < 192 * 16) { int r = i >> 4, c = i & 15; w2p_g[i] = (c < 10) ? mg_w2[r * 10 + c] : 0.0f; }
    if (i < 16) {
        b2p_n[i] = (i < 10) ? mn_b2[i] : 0.0f;
        b2p_g[i] = (i < 10) ? mg_b2[i] : 0.0f;
    }
}

extern "C" void kernel_launch(void* const* d_in, const int* in_sizes, int n_in,
                              void* d_out, int out_size, void* d_ws, size_t ws_size,
                              hipStream_t stream) {
    (void)n_in; (void)out_size; (void)ws_size;
    const float* feature = (const float*)d_in[0];
    const float* W1 = (const float*)d_in[1];  const float* b1 = (const float*)d_in[2];
    const float* W2 = (const float*)d_in[3];  const float* b2 = (const float*)d_in[4];
    const float* W3 = (const float*)d_in[5];  const float* b3 = (const float*)d_in[6];
    const float* Ws = (const float*)d_in[7];  const float* bs = (const float*)d_in[8];
    const float* mg_w0 = (const float*)d_in[9];  const float* mg_b0 = (const float*)d_in[10];
    const float* mg_w1 = (const float*)d_in[11]; const float* mg_b1 = (const float*)d_in[12];
    const float* mg_w2 = (const float*)d_in[13]; const float* mg_b2 = (const float*)d_in[14];
    const float* mn_w0 = (const float*)d_in[15]; const float* mn_b0 = (const float*)d_in[16];
    const float* mn_w1 = (const float*)d_in[17]; const float* mn_b1 = (const float*)d_in[18];
    const float* mn_w2 = (const float*)d_in[19]; const float* mn_b2 = (const float*)d_in[20];
    const int* src = (const int*)d_in[21];
    const int* dst = (const int*)d_in[22];

    const int N  = in_sizes[0] / 128;
    const int E  = in_sizes[21];
    const int Bg = N / 1024;

    // ---- workspace layout (floats) ----
    float* ws       = (float*)d_ws;
    float* norm_out = ws;                              // N
    float* norm_in  = norm_out + (size_t)N;            // N
    float* agg      = norm_in + (size_t)N;             // N*128
    float* out3     = agg + (size_t)N * 128;           // N*384
    float* s0buf    = out3 + (size_t)N * 384;          // N
    float* sagg     = s0buf + (size_t)N;               // N
    float* score    = sagg + (size_t)N;                // N
    float* hg3      = score + (size_t)N;               // Bg*768
    float* gh0      = hg3 + (size_t)Bg * 768;          // Bg*384
    float* gh1      = gh0 + (size_t)Bg * 384;          // Bg*192
    float* w2p_n    = gh1 + (size_t)Bg * 192;          // 96*16
    float* b2p_n    = w2p_n + 96 * 16;                 // 16
    float* w2p_g    = b2p_n + 16;                      // 192*16
    float* b2p_g    = w2p_g + 192 * 16;                // 16
    int*   perm     = (int*)(b2p_g + 16);              // Bg*1024

    // ---- output layout ----
    float* outp         = (float*)d_out;
    float* o_scores     = outp;                            // Bg*10
    float* o_scores_com = o_scores + (size_t)Bg * 10;      // Bg*10
    float* o_hg3_com    = o_scores_com + (size_t)Bg * 10;  // Bg*768
    float* o_pred       = o_hg3_com + (size_t)Bg * 768;    // N*10
    float* o_nscore     = o_pred + (size_t)N * 10;         // N

    // degrees -> norms
    zero_kernel<<<256, 256, 0, stream>>>(norm_out, (size_t)2 * N);
    deg_kernel<<<(E + 255) / 256, 256, 0, stream>>>(src, dst, norm_out, norm_in, E);
    norm_kernel<<<(2 * N + 255) / 256, 256, 0, stream>>>(norm_out, (size_t)2 * N);
    pad_kernel<<<(192 * 16 + 255) / 256, 256, 0, stream>>>(mn_w2, mn_b2, mg_w2, mg_b2,
                                                           w2p_n, b2p_n, w2p_g, b2p_g);

    // 3 graph-conv layers
    const float* Xl[3]  = {feature, out3, out3 + 128};
    const int    ldal[3] = {128, 384, 384};
    const float* Wl[3]  = {W1, W2, W3};
    const float* bl[3]  = {b1, b2, b3};
    for (int l = 0; l < 3; ++l) {
        zero_kernel<<<2048, 256, 0, stream>>>(agg, (size_t)N * 128);
        gather_scatter<<<(E + 7) / 8, 256, 0, stream>>>(Xl[l], ldal[l], src, dst,
                                                        norm_out, agg, E);
        dim3 gg(N / 16 / 8, 128 / 16);
        gemm_wmma<<<gg, 256, 0, stream>>>(agg, 128, Wl[l], 128, bl[l], norm_in,
                                          out3 + l * 128, 384, 128, 128, 1);
    }

    // node classifier (fused, WMMA)
    node_mlp_fused<<<N / 16, 256, 0, stream>>>(out3, mn_w0, mn_b0, mn_w1, mn_b1,
                                               w2p_n, b2p_n, o_pred);

    // score path
    s0_kernel<<<(N + 7) / 8, 256, 0, stream>>>(out3, Ws, s0buf, N);
    zero_kernel<<<64, 256, 0, stream>>>(sagg, (size_t)N);
    score_edge<<<(E + 255) / 256, 256, 0, stream>>>(src, dst, s0buf, norm_out, sagg, E);
    score_fin<<<(N + 255) / 256, 256, 0, stream>>>(sagg, norm_in, bs, score, N);

    // top-k + pooling
    sort_kernel<<<Bg, 512, 0, stream>>>(score, perm);
    pool_kernel<<<dim3(Bg, 2), 384, 0, stream>>>(out3, score, perm, hg3, o_hg3_com);

    // graph classifier (dis then com), M=64 -> 4 waves per block
    dim3 g0(1, 384 / 16), g1(1, 192 / 16), g2(1, 1);
    gemm_wmma<<<g0, 128, 0, stream>>>(hg3, 768, mg_w0, 384, mg_b0, nullptr, gh0, 384, 384, 768, 1);
    gemm_wmma<<<g1, 128, 0, stream>>>(gh0, 384, mg_w1, 192, mg_b1, nullptr, gh1, 192, 192, 384, 1);
    gemm_wmma<<<g2, 128, 0, stream>>>(gh1, 192, w2p_g, 16, b2p_g, nullptr, o_scores, 10, 10, 192, 0);
    gemm_wmma<<<g0, 128, 0, stream>>>(o_hg3_com, 768, mg_w0, 384, mg_b0, nullptr, gh0, 384, 384, 768, 1);
    gemm_wmma<<<g1, 128, 0, stream>>>(gh0, 384, mg_w1, 192, mg_b1, nullptr, gh1, 192, 192, 384, 1);
    gemm_wmma<<<g2, 128, 0, stream>>>(gh1, 192, w2p_g, 16, b2p_g, nullptr, o_scores_com, 10, 10, 192, 0);

    // global softmax over scores
    softmax_kernel<<<1, 1024, 0, stream>>>(score, o_nscore, N);
}